// dcn_conv_relu_60344290509582
// MI455X (gfx1250) — compile-verified
//
#include <hip/hip_runtime.h>
#include <math.h>

// DCNv2 fused, gfx1250 wave32:
//   stage 0: x patch -> LDS
//   stage 1: offset(18)+mod(9) conv as 32x32 WMMA f32 GEMM, K split over 8 waves
//   stage 2: bilinear corner indices + (valid * 2*sigmoid(mod))-premultiplied weights
//   stage 3: per-tap gather into ping-pong LDS panel + 64x32 WMMA f32 GEMM
//   epilogue: bias + ReLU

typedef __attribute__((ext_vector_type(2))) float v2f;
typedef __attribute__((ext_vector_type(8))) float v8f;

#define HH 192
#define WW 192
#define CIN 64
#define COUT 64
#define KK2 9
#define TPIX 32          // pixels per block tile (one row segment)
#define KTOT 576         // CIN * KK2
#define VSTR 48          // valp pixel-row stride (keeps lane halves in disjoint banks)
#define VBUF (CIN * VSTR) // 3072 floats per sample-panel buffer

__launch_bounds__(256)
__global__ void dcn_fused_kernel(const float* __restrict__ x,
                                 const float* __restrict__ w_off,
                                 const float* __restrict__ b_off,
                                 const float* __restrict__ w_mod,
                                 const float* __restrict__ b_mod,
                                 const float* __restrict__ w,
                                 const float* __restrict__ b,
                                 float* __restrict__ out) {
    // smemA: stage 0/1 = x patch (64*3*34 = 6528 floats)
    //        stage 3   = two sample panels (2 * 3072 = 6144 floats)
    __shared__ float smemA[CIN * 3 * 34];
    __shared__ float offmodA[32 * 36];         // stage-1 partial (taps 0..4)
    __shared__ float offmodB[32 * 36];         // stage-1 partial (taps 5..8)
    __shared__ int   sidx[KK2 * TPIX * 4];     // clamped corner flat idx per (tap,pix)
    __shared__ float swt [KK2 * TPIX * 4];     // corner weights (premult valid*mod)

    const int tid  = threadIdx.x;
    const int lane = tid & 31;
    const int wv   = tid >> 5;
    const int half = lane >> 4;
    const int l16  = lane & 15;

    const int bid = blockIdx.x;
    const int tx  = bid % (WW / TPIX);
    const int oy  = (bid / (WW / TPIX)) % HH;
    const int n   = bid / ((WW / TPIX) * HH);
    const int ox0 = tx * TPIX;

    const float* xn = x + (size_t)n * CIN * HH * WW;

    // ---------------- stage 0: load zero-padded x patch ----------------
    for (int i = tid; i < CIN * 3 * 34; i += 256) {
        int c   = i / 102;
        int r   = (i % 102) / 34;
        int col = i % 34;
        int gy = oy - 1 + r;
        int gx = ox0 - 1 + col;
        float v = 0.0f;
        if (gy >= 0 && gy < HH && gx >= 0 && gx < WW)
            v = xn[(c * HH + gy) * WW + gx];
        smemA[i] = v;
    }
    __syncthreads();

    // ------- stage 1: offset/mod conv, 32x32 WMMA GEMM, K split over waves -----
    {
        const int wgrp = wv >> 2;              // 0: taps 0..4, 1: taps 5..8
        const int wsub = wv & 3;
        const int m_base = (wsub >> 1) * 16;
        const int p_base = (wsub & 1) * 16;
        const int m   = m_base + l16;          // A row this lane reads
        const int pix = p_base + l16;          // B col this lane reads

        // branch-free per-lane A row pointer + scale (hoisted out of hot loop)
        const float* arow = w_off;             // dummy row for m >= 27
        float ascale = 0.0f;
        if (m < 18)      { arow = w_off + m * KTOT;        ascale = 1.0f; }
        else if (m < 27) { arow = w_mod + (m - 18) * KTOT; ascale = 1.0f; }

        const int tap_lo = wgrp ? 5 : 0;
        const int tap_hi = wgrp ? 9 : 5;
        v8f acc = {};
        for (int tap = tap_lo; tap < tap_hi; ++tap) {
            const int ky = tap / 3, kx = tap % 3;
            const int pcol = ky * 34 + pix + kx;
            v2f areg[16], breg[16];
#pragma unroll
            for (int ks = 0; ks < 16; ++ks) {
                const int ca = ks * 4 + 2 * half;
                areg[ks].x = ascale * arow[ca * 9 + tap];
                areg[ks].y = ascale * arow[(ca + 1) * 9 + tap];
            }
#pragma unroll
            for (int ks = 0; ks < 16; ++ks) {
                const int ca = ks * 4 + 2 * half;
                breg[ks].x = smemA[ca * 102 + pcol];
                breg[ks].y = smemA[(ca + 1) * 102 + pcol];
            }
#pragma unroll
            for (int ks = 0; ks < 16; ++ks)
                acc = __builtin_amdgcn_wmma_f32_16x16x4_f32(
                        false, areg[ks], false, breg[ks], (short)0, acc, false, false);
        }
        // D layout: VGPR j -> row m_base+j (lanes0-15) / m_base+j+8 (lanes16-31)
        float* om = wgrp ? offmodB : offmodA;
#pragma unroll
        for (int j = 0; j < 8; ++j)
            om[(m_base + j + 8 * half) * 36 + pix] = acc[j];
    }
    __syncthreads();

    // ---------------- stage 2: bilinear params per (tap, pix) -----------------
    for (int it = tid; it < KK2 * TPIX; it += 256) {
        const int pix = it & 31;
        const int tap = it >> 5;
        const int ky = tap / 3, kx = tap % 3;
        float dy = offmodA[(2 * tap) * 36 + pix]     + offmodB[(2 * tap) * 36 + pix]
                 + b_off[2 * tap];
        float dx = offmodA[(2 * tap + 1) * 36 + pix] + offmodB[(2 * tap + 1) * 36 + pix]
                 + b_off[2 * tap + 1];
        float mz = offmodA[(18 + tap) * 36 + pix]    + offmodB[(18 + tap) * 36 + pix]
                 + b_mod[tap];
        float modv = 2.0f / (1.0f + expf(-mz));

        float py = dy + (float)(oy - 1 + ky);
        float px = dx + (float)(ox0 + pix - 1 + kx);
        float y0f = floorf(py), x0f = floorf(px);
        float ly = py - y0f,    lx = px - x0f;
        int y0 = (int)y0f, x0 = (int)x0f;
        int y1 = y0 + 1,   x1 = x0 + 1;
        bool vy0 = (y0 >= 0) && (y0 < HH);
        bool vy1 = (y1 >= 0) && (y1 < HH);
        bool vx0 = (x0 >= 0) && (x0 < WW);
        bool vx1 = (x1 >= 0) && (x1 < WW);
        int y0c = min(max(y0, 0), HH - 1);
        int y1c = min(max(y1, 0), HH - 1);
        int x0c = min(max(x0, 0), WW - 1);
        int x1c = min(max(x1, 0), WW - 1);

        const int pbse = it * 4;
        sidx[pbse + 0] = y0c * WW + x0c;
        sidx[pbse + 1] = y0c * WW + x1c;
        sidx[pbse + 2] = y1c * WW + x0c;
        sidx[pbse + 3] = y1c * WW + x1c;
        swt[pbse + 0] = (1.0f - ly) * (1.0f - lx) * modv * ((vy0 && vx0) ? 1.0f : 0.0f);
        swt[pbse + 1] = (1.0f - ly) * lx          * modv * ((vy0 && vx1) ? 1.0f : 0.0f);
        swt[pbse + 2] = ly * (1.0f - lx)          * modv * ((vy1 && vx0) ? 1.0f : 0.0f);
        swt[pbse + 3] = ly * lx                   * modv * ((vy1 && vx1) ? 1.0f : 0.0f);
    }
    __syncthreads();

    // ------- stage 3: per-tap gather (ping-pong panel) + 64x32 WMMA GEMM ------
    const int m_base = (wv & 3) * 16;    // 4 M-tiles
    const int p_base = (wv >> 2) * 16;   // 2 N-tiles
    const int o    = m_base + l16;       // A row (out channel) this lane reads
    const int pixw = p_base + l16;       // B col this lane reads
    const float* wrow = w + o * KTOT;
    v8f acc = {};

    const int spix = tid & 31;           // sampling: pixel handled by this thread
    const int c8   = (tid >> 5) * 8;     // sampling: 8 channels per thread

    for (int tap = 0; tap < KK2; ++tap) {
        float* vb = smemA + ((tap & 1) ? VBUF : 0);

        const int pbse = (tap * TPIX + spix) * 4;
        const int i00 = sidx[pbse + 0], i01 = sidx[pbse + 1];
        const int i10 = sidx[pbse + 2], i11 = sidx[pbse + 3];
        const float w00 = swt[pbse + 0], w01 = swt[pbse + 1];
        const float w10 = swt[pbse + 2], w11 = swt[pbse + 3];

        // gather + blend 8 channels for this pixel into the current panel
#pragma unroll
        for (int j = 0; j < 8; ++j) {
            const int c = c8 + j;
            const float* xc = xn + (size_t)c * (HH * WW);
            vb[c * VSTR + spix] = w00 * xc[i00] + w01 * xc[i01]
                                + w10 * xc[i10] + w11 * xc[i11];
        }

        // hoist all A fragments for this tap (latency overlaps the barrier)
        v2f areg[16];
#pragma unroll
        for (int ks = 0; ks < 16; ++ks) {
            const int ca = ks * 4 + 2 * half;
            areg[ks].x = wrow[ca * 9 + tap];
            areg[ks].y = wrow[(ca + 1) * 9 + tap];
        }

        __syncthreads();   // panel ready; implicit s_wait_dscnt makes ping-pong safe

        v2f breg[16];
#pragma unroll
        for (int ks = 0; ks < 16; ++ks) {
            const int ca = ks * 4 + 2 * half;
            breg[ks].x = vb[ca * VSTR + pixw];
            breg[ks].y = vb[(ca + 1) * VSTR + pixw];
        }
#pragma unroll
        for (int ks = 0; ks < 16; ++ks)
            acc = __builtin_amdgcn_wmma_f32_16x16x4_f32(
                    false, areg[ks], false, breg[ks], (short)0, acc, false, false);
    }

    // ---------------- epilogue: bias + ReLU, store ----------------------------
#pragma unroll
    for (int j = 0; j < 8; ++j) {
        const int oo = m_base + j + 8 * half;
        float r = acc[j] + b[oo];
        r = fmaxf(r, 0.0f);
        out[(((size_t)n * COUT + oo) * HH + oy) * WW + (ox0 + p_base + l16)] = r;
    }
}

extern "C" void kernel_launch(void* const* d_in, const int* in_sizes, int n_in,
                              void* d_out, int out_size, void* d_ws, size_t ws_size,
                              hipStream_t stream) {
    const float* x     = (const float*)d_in[0];
    const float* w_off = (const float*)d_in[1];
    const float* b_off = (const float*)d_in[2];
    const float* w_mod = (const float*)d_in[3];
    const float* b_mod = (const float*)d_in[4];
    const float* w     = (const float*)d_in[5];
    const float* b     = (const float*)d_in[6];
    float* out = (float*)d_out;

    const int N = in_sizes[0] / (CIN * HH * WW);   // 4
    const int blocks = N * HH * (WW / TPIX);       // 4 * 192 * 6 = 4608
    dcn_fused_kernel<<<blocks, 256, 0, stream>>>(x, w_off, b_off, w_mod, b_mod,
                                                 w, b, out);
}